// GCN_86414741995856
// MI455X (gfx1250) — compile-verified
//
#include <hip/hip_runtime.h>
#include <hip/hip_bf16.h>
#include <math.h>

// ---------------------------------------------------------------------------
// 2-layer GCN forward for MI455X (gfx1250, wave32).
//  GEMM1 uses V_WMMA_F32_16X16X4_F32 (full fp32 precision, matches reference).
//  Edge aggregation uses L2-resident float atomics (h1 = 6.4MB << 192MB L2).
// ---------------------------------------------------------------------------

typedef float v2f __attribute__((ext_vector_type(2)));
typedef float v8f __attribute__((ext_vector_type(8)));

#define NF   512   // input features
#define NH   16    // hidden
#define NC   40    // classes

// ---------------------------------------------------------------------------
// 0) init: deg=1 (self loop), agg1=0, logits seeded with b2
// ---------------------------------------------------------------------------
__global__ void gcn_init_kernel(float* __restrict__ deg,
                                float* __restrict__ agg1,
                                float* __restrict__ logits,
                                const float* __restrict__ b2, int N) {
    int i = blockIdx.x * blockDim.x + threadIdx.x;
    int total = N * NC;
    if (i >= total) return;
    logits[i] = b2[i % NC];
    if (i < N * NH) agg1[i] = 0.0f;
    if (i < N)      deg[i]  = 1.0f;   // self-loop contribution
}

// ---------------------------------------------------------------------------
// 1) in-degree of dst (real edges)
// ---------------------------------------------------------------------------
__global__ void gcn_deg_kernel(const int* __restrict__ dst,
                               float* __restrict__ deg, int E) {
    int e = blockIdx.x * blockDim.x + threadIdx.x;
    if (e >= E) return;
    atomicAdd(&deg[dst[e]], 1.0f);
}

// deg -> rsqrt(deg) in place
__global__ void gcn_dinv_kernel(float* __restrict__ deg, int N) {
    int i = blockIdx.x * blockDim.x + threadIdx.x;
    if (i >= N) return;
    float d = deg[i];
    deg[i] = (d > 0.0f) ? rsqrtf(d) : 0.0f;
}

// ---------------------------------------------------------------------------
// 2) GEMM1: h1 = X @ W1 via V_WMMA_F32_16X16X4_F32.
//    256 threads = 8 waves; each wave computes one 16x16 output tile
//    (16 rows of X, all 16 hidden cols), sweeping K=512 in steps of 4.
//    W1 (512x16 = 32KB) staged in LDS once per block.
// ---------------------------------------------------------------------------
__global__ void gcn_gemm1_wmma_kernel(const float* __restrict__ X,
                                      const float* __restrict__ W1,
                                      float* __restrict__ h1, int N) {
    __shared__ float w1s[NF * NH];            // 32 KB
    for (int i = threadIdx.x; i < NF * NH; i += blockDim.x)
        w1s[i] = W1[i];
    __syncthreads();

    const int wid   = threadIdx.x >> 5;       // wave in block (0..7)
    const int lane  = threadIdx.x & 31;
    const int rowBase = (blockIdx.x * 8 + wid) * 16;

    const int nsel  = lane & 15;              // N-column / M-row selector
    const int khalf = (lane >> 4) << 1;       // 0 for lanes 0-15, 2 for 16-31

    // Clamp row so every lane issues valid loads: EXEC must stay all-ones
    // across the WMMA (ISA 7.12 restriction). Out-of-range results are
    // discarded at the store.
    int row = rowBase + nsel;
    if (row >= N) row = N - 1;
    const float* xrow = X + (size_t)row * NF;

    v8f acc = {};
    #pragma unroll 8
    for (int k0 = 0; k0 < NF; k0 += 4) {
        const int ka = k0 + khalf;            // even -> 8B aligned
        // A fragment: 16x4 f32, lane holds A[M=nsel][ka], A[M=nsel][ka+1]
        float2 a2 = *reinterpret_cast<const float2*>(xrow + ka);
        v2f A; A.x = a2.x; A.y = a2.y;
        // B fragment: 4x16 f32, lane holds W1[ka][nsel], W1[ka+1][nsel]
        v2f B; B.x = w1s[ka * NH + nsel];
               B.y = w1s[(ka + 1) * NH + nsel];
        acc = __builtin_amdgcn_wmma_f32_16x16x4_f32(
            /*neg_a=*/false, A, /*neg_b=*/false, B,
            /*c_mod=*/(short)0, acc, /*reuse_a=*/false, /*reuse_b=*/false);
    }

    // C/D layout: VGPR r holds M = r (+8 for upper lane half), N = nsel.
    const int mBase = (lane >> 4) << 3;
    #pragma unroll
    for (int r = 0; r < 8; ++r) {
        int outRow = rowBase + mBase + r;
        if (outRow < N) h1[(size_t)outRow * NH + nsel] = acc[r];
    }
}

// ---------------------------------------------------------------------------
// 3) edge aggregation, 16-wide: agg1[dst] += h1[src] * dinv[src]*dinv[dst]
//    lane j of each 16-lane group handles feature j -> coalesced 64B gathers.
//    Edges [E, E+N) are the self loops.
// ---------------------------------------------------------------------------
__global__ void gcn_agg16_kernel(const int* __restrict__ src,
                                 const int* __restrict__ dst,
                                 const float* __restrict__ dinv,
                                 const float* __restrict__ h1,
                                 float* __restrict__ agg1,
                                 int E, int N) {
    long long tid = (long long)blockIdx.x * blockDim.x + threadIdx.x;
    long long total = (long long)(E + N) * NH;
    if (tid >= total) return;
    int e = (int)(tid >> 4);
    int j = (int)(tid & 15);
    int s, d;
    if (e < E) { s = src[e]; d = dst[e]; }
    else       { s = d = e - E; }
    float w = dinv[s] * dinv[d];
    atomicAdd(&agg1[(size_t)d * NH + j], h1[(size_t)s * NH + j] * w);
}

// h = relu(agg1 + b1), in place
__global__ void gcn_relu_bias_kernel(float* __restrict__ h,
                                     const float* __restrict__ b1, int N) {
    int i = blockIdx.x * blockDim.x + threadIdx.x;
    if (i >= N * NH) return;
    float v = h[i] + b1[i & 15];
    h[i] = v > 0.0f ? v : 0.0f;
}

// ---------------------------------------------------------------------------
// 4) GEMM2: h2 = h @ W2  (K=16, 40 cols; 128 MFLOP -> scalar FMAs, W2 in LDS)
// ---------------------------------------------------------------------------
__global__ void gcn_gemm2_kernel(const float* __restrict__ h,
                                 const float* __restrict__ W2,
                                 float* __restrict__ h2, int N) {
    __shared__ float w2s[NH * NC];            // 2.5 KB
    for (int i = threadIdx.x; i < NH * NC; i += blockDim.x)
        w2s[i] = W2[i];
    __syncthreads();

    int n = blockIdx.x * blockDim.x + threadIdx.x;
    if (n >= N) return;
    float hr[NH];
    #pragma unroll
    for (int k = 0; k < NH; ++k) hr[k] = h[(size_t)n * NH + k];
    #pragma unroll 4
    for (int c = 0; c < NC; ++c) {
        float s = 0.0f;
        #pragma unroll
        for (int k = 0; k < NH; ++k) s = fmaf(hr[k], w2s[k * NC + c], s);
        h2[(size_t)n * NC + c] = s;
    }
}

// ---------------------------------------------------------------------------
// 5) edge aggregation, 40-wide, accumulating into logits (pre-seeded with b2)
// ---------------------------------------------------------------------------
__global__ void gcn_agg40_kernel(const int* __restrict__ src,
                                 const int* __restrict__ dst,
                                 const float* __restrict__ dinv,
                                 const float* __restrict__ h2,
                                 float* __restrict__ logits,
                                 int E, int N) {
    long long tid = (long long)blockIdx.x * blockDim.x + threadIdx.x;
    long long total = (long long)(E + N) * NC;
    if (tid >= total) return;
    int e = (int)(tid / NC);
    int c = (int)(tid % NC);
    int s, d;
    if (e < E) { s = src[e]; d = dst[e]; }
    else       { s = d = e - E; }
    float w = dinv[s] * dinv[d];
    atomicAdd(&logits[(size_t)d * NC + c], h2[(size_t)s * NC + c] * w);
}

// ---------------------------------------------------------------------------
// 6) log_softmax(logits) -> ls ; logits already hold final values
// ---------------------------------------------------------------------------
__global__ void gcn_logsoftmax_kernel(const float* __restrict__ logits,
                                      float* __restrict__ ls, int N) {
    int n = blockIdx.x * blockDim.x + threadIdx.x;
    if (n >= N) return;
    const float* lr = logits + (size_t)n * NC;
    float m = -INFINITY;
    #pragma unroll 4
    for (int c = 0; c < NC; ++c) m = fmaxf(m, lr[c]);
    float s = 0.0f;
    #pragma unroll 4
    for (int c = 0; c < NC; ++c) s += expf(lr[c] - m);
    float lse = m + logf(s);
    #pragma unroll 4
    for (int c = 0; c < NC; ++c) ls[(size_t)n * NC + c] = lr[c] - lse;
}

// ---------------------------------------------------------------------------
extern "C" void kernel_launch(void* const* d_in, const int* in_sizes, int n_in,
                              void* d_out, int out_size, void* d_ws, size_t ws_size,
                              hipStream_t stream) {
    const float* X   = (const float*)d_in[0];
    const int*   ei  = (const int*)d_in[1];   // edge_index [2, E]
    const float* W1  = (const float*)d_in[2];
    const float* b1  = (const float*)d_in[3];
    const float* W2  = (const float*)d_in[4];
    const float* b2  = (const float*)d_in[5];

    const int N = in_sizes[0] / NF;           // 100000
    const int E = in_sizes[1] / 2;            // 3200000
    const int* src = ei;
    const int* dst = ei + E;

    // workspace layout (floats): dinv | h1 | agg1 | h2   (~29.2 MB total)
    float* ws   = (float*)d_ws;
    float* dinv = ws;
    float* h1   = dinv + N;
    float* agg1 = h1 + (size_t)N * NH;
    float* h2   = agg1 + (size_t)N * NH;

    float* ls     = (float*)d_out;                 // [N, 40] log_softmax
    float* logits = (float*)d_out + (size_t)N * NC;// [N, 40] logits

    const int B = 256;

    // 0) init
    {
        long long tot = (long long)N * NC;
        gcn_init_kernel<<<(unsigned)((tot + B - 1) / B), B, 0, stream>>>(
            dinv, agg1, logits, b2, N);
    }
    // 1) degree + rsqrt
    gcn_deg_kernel<<<(E + B - 1) / B, B, 0, stream>>>(dst, dinv, E);
    gcn_dinv_kernel<<<(N + B - 1) / B, B, 0, stream>>>(dinv, N);
    // 2) GEMM1 via fp32 WMMA: 8 waves/block, 16 rows/wave -> 128 rows/block
    gcn_gemm1_wmma_kernel<<<(N + 127) / 128, B, 0, stream>>>(X, W1, h1, N);
    // 3) aggregate hidden
    {
        long long tot = (long long)(E + N) * NH;
        gcn_agg16_kernel<<<(unsigned)((tot + B - 1) / B), B, 0, stream>>>(
            src, dst, dinv, h1, agg1, E, N);
    }
    gcn_relu_bias_kernel<<<((long long)N * NH + B - 1) / B, B, 0, stream>>>(agg1, b1, N);
    // 4) GEMM2
    gcn_gemm2_kernel<<<(N + B - 1) / B, B, 0, stream>>>(agg1, W2, h2, N);
    // 5) aggregate logits (into d_out, seeded with b2)
    {
        long long tot = (long long)(E + N) * NC;
        gcn_agg40_kernel<<<(unsigned)((tot + B - 1) / B), B, 0, stream>>>(
            src, dst, dinv, h2, logits, E, N);
    }
    // 6) log_softmax
    gcn_logsoftmax_kernel<<<(N + B - 1) / B, B, 0, stream>>>(logits, ls, N);
}